// FilterSum_12833362280592
// MI455X (gfx1250) — compile-verified
//
#include <hip/hip_runtime.h>

// CDNA5 / gfx1250, wave32. out[b,s,t] = sum_d in[b,t,s,d] * aff[d,t]
// B=8, T=128, S=256, D=1024. Memory-bound: stream 1 GiB of input once
// (HBM floor ~46 us @ 23.3 TB/s). Reduction engine: V_WMMA_F32_16X16X4_F32
// with broadcast B-columns (full fp32 precision; the 16x redundant columns
// are free — HBM is the limiter). Input loads / output stores are marked
// non-temporal so the one-pass 1 GiB stream doesn't churn the 192 MB L2,
// which keeps the shared 512 KB affine matrix resident.

typedef float v2f __attribute__((ext_vector_type(2)));
typedef float v8f __attribute__((ext_vector_type(8)));

#define B_ 8
#define T_ 128
#define S_ 256
#define D_ 1024

__global__ __launch_bounds__(256, 1) void filtersum_wmma(
    const float* __restrict__ in,   // [B, T, S, D]
    const float* __restrict__ aff,  // [D, T]
    float* __restrict__ out)        // [B, S, T]
{
    __shared__ float wcol[D_];

    const int blk = blockIdx.x;       // 0 .. B*T-1
    const int b = blk / T_;
    const int t = blk - b * T_;

    // Stage affine column t into LDS (4 KB). Strided gather, but affine is
    // only 512 KB total -> stays L2 resident across all 1024 blocks.
    for (int d = threadIdx.x; d < D_; d += 256) {
        wcol[d] = aff[(size_t)d * T_ + t];
    }
    __syncthreads();

    const int lane = threadIdx.x & 31;
    const int wave = threadIdx.x >> 5;   // 0..7
    const int row  = lane & 15;          // M index within 16-row s-tile
    const int half = lane >> 4;          // 0 -> K={0,1}, 1 -> K={2,3}

    const size_t bt_base = ((size_t)(b * T_) + t) * (size_t)(S_ * D_);

    // Each wave owns two 16-row s-tiles: 8 waves * 2 * 16 = 256 = S.
    const int s0a = (wave * 2 + 0) * 16;
    const int s0b = (wave * 2 + 1) * 16;

    // A-fragment layout (32-bit A 16x4): lane l<16 holds A[l, 0..1],
    // lane l+16 holds A[l, 2..3] -> per-lane contiguous 8-byte load.
    const v2f* rowA = (const v2f*)(in + bt_base + (size_t)(s0a + row) * D_ + 2 * half);
    const v2f* rowB = (const v2f*)(in + bt_base + (size_t)(s0b + row) * D_ + 2 * half);

    v8f accA = {0.f, 0.f, 0.f, 0.f, 0.f, 0.f, 0.f, 0.f};
    v8f accB = {0.f, 0.f, 0.f, 0.f, 0.f, 0.f, 0.f, 0.f};

#pragma unroll 8
    for (int d0 = 0; d0 < D_; d0 += 4) {
        // B-fragment (4x16, broadcast over the 16 columns):
        // VGPR0 = B[half, n]   = w[d0 + half]
        // VGPR1 = B[2+half, n] = w[d0 + half + 2]
        v2f bw;
        bw.x = wcol[d0 + half];
        bw.y = wcol[d0 + half + 2];

        // Non-temporal: single-pass stream, no cross-wave reuse.
        v2f aA = __builtin_nontemporal_load(rowA + (d0 >> 1));
        v2f aB = __builtin_nontemporal_load(rowB + (d0 >> 1));

        accA = __builtin_amdgcn_wmma_f32_16x16x4_f32(
            /*neg_a=*/false, aA, /*neg_b=*/false, bw,
            /*c_mod=*/(short)0, accA, /*reuse_a=*/false, /*reuse_b=*/false);
        accB = __builtin_amdgcn_wmma_f32_16x16x4_f32(
            /*neg_a=*/false, aB, /*neg_b=*/false, bw,
            /*c_mod=*/(short)0, accB, /*reuse_a=*/false, /*reuse_b=*/false);
    }

    // D[m,n] is identical across n (B columns were broadcast).
    // C/D layout: lane 0 (col 0) holds rows 0..7 in acc[0..7];
    // lane 16 holds rows 8..15. Two lanes per wave write 16 results/tile.
    if ((lane & 15) == 0) {
        float* oa = out + ((size_t)b * S_ + (size_t)(s0a + half * 8)) * T_ + t;
        float* ob = out + ((size_t)b * S_ + (size_t)(s0b + half * 8)) * T_ + t;
#pragma unroll
        for (int v = 0; v < 8; ++v) {
            __builtin_nontemporal_store(accA[v], oa + (size_t)v * T_);
            __builtin_nontemporal_store(accB[v], ob + (size_t)v * T_);
        }
    }
}

extern "C" void kernel_launch(void* const* d_in, const int* in_sizes, int n_in,
                              void* d_out, int out_size, void* d_ws, size_t ws_size,
                              hipStream_t stream) {
    (void)in_sizes; (void)n_in; (void)d_ws; (void)ws_size; (void)out_size;
    const float* in  = (const float*)d_in[0];   // [B,T,S,D] fp32
    const float* aff = (const float*)d_in[1];   // [D,T]     fp32
    float* out = (float*)d_out;                 // [B,S,T]   fp32

    dim3 grid(B_ * T_);   // one block per (b, t)
    dim3 block(256);      // 8 waves (wave32)
    hipLaunchKernelGGL(filtersum_wmma, grid, block, 0, stream, in, aff, out);
}